// QuantumAutoEncoder_44581760532621
// MI455X (gfx1250) — compile-verified
//
#include <hip/hip_runtime.h>
#include <hip/hip_bf16.h>

typedef __attribute__((ext_vector_type(2))) float v2f;
typedef __attribute__((ext_vector_type(4))) float v4f;
typedef __attribute__((ext_vector_type(8))) float v8f;
typedef __attribute__((ext_vector_type(4))) unsigned int u32x4;
typedef __attribute__((ext_vector_type(4))) int i32x4;
typedef __attribute__((ext_vector_type(8))) int i32x8;

#define N_QUBITS 6
#define DIM 64

#if __has_builtin(__builtin_amdgcn_tensor_load_to_lds) && \
    __has_builtin(__builtin_amdgcn_s_wait_tensorcnt)
#define USE_TDM 1
#else
#define USE_TDM 0
#endif

#if USE_TDM
// Issue a TDM 2D tile load (tile == whole tensor) of f32 data into LDS.
// D# packing per CDNA5 ISA.
__device__ inline void tdm_load_2d_f32(unsigned int lds_off, const void* gptr,
                                       unsigned int dim0, unsigned int dim1) {
    const unsigned long long ga = (unsigned long long)(uintptr_t)gptr;
    u32x4 g0;
    g0[0] = 1u;                                   // count=1, user-mode, no gather
    g0[1] = lds_off;                              // lds_addr (bytes)
    g0[2] = (unsigned int)ga;                     // global_addr[31:0]
    g0[3] = (unsigned int)((ga >> 32) & 0x01FFFFFFu) | (2u << 30); // addr[56:32] | type=2
    i32x8 g1;
    g1[0] = (int)(2u << 16);                      // wg_mask=0, data_size=4B
    g1[1] = (int)((dim0 & 0xFFFFu) << 16);        // tensor_dim0[15:0]
    g1[2] = (int)(((dim0 >> 16) & 0xFFFFu) | ((dim1 & 0xFFFFu) << 16)); // dim0 hi|dim1 lo
    g1[3] = (int)(((dim1 >> 16) & 0xFFFFu) | ((dim0 & 0xFFFFu) << 16)); // dim1 hi|tile_dim0
    g1[4] = (int)(dim1 & 0xFFFFu);                // tile_dim1 | tile_dim2=0
    g1[5] = (int)dim0;                            // tensor_dim0_stride lo
    g1[6] = 0;                                    // stride0 hi | stride1 lo
    g1[7] = 0;
    const i32x4 z4 = {0, 0, 0, 0};
#if __clang_major__ >= 23
    const i32x8 z8 = {0, 0, 0, 0, 0, 0, 0, 0};
    __builtin_amdgcn_tensor_load_to_lds(g0, g1, z4, z4, z8, 0);
#else
    __builtin_amdgcn_tensor_load_to_lds(g0, g1, z4, z4, 0);
#endif
}
#endif

// ---------------------------------------------------------------------------
// Kernel 1: build the 64x64 circuit unitaries from enc_w / dec_w.
// Thread t < 64 : encoder column t -> written in WMMA-B fragment-packed order:
//   UP[((k>>1)*64 + n)*4 + {0,1,2,3}] = {Re[k][n], Re[k+1][n], Im[k][n], Im[k+1][n]}
// so the GEMM kernel fetches both (re,im) B fragments with one ds_load_b128.
// Thread t >= 64: decoder column t-64; only columns 0..2 stored (decoder input
// state has only 3 non-zero amplitudes): Dec[j*64+n] re, Dec[192+j*64+n] im.
// ---------------------------------------------------------------------------
__global__ __launch_bounds__(128)
void qae_build_U(const float* __restrict__ encw, const float* __restrict__ decw,
                 float* __restrict__ UP, float* __restrict__ DecG) {
    __shared__ float vre[128 * 65];
    __shared__ float vim[128 * 65];
    const int t   = threadIdx.x;      // 0..127
    const int col = t & 63;
    const float* w = (t < 64) ? encw : decw;
    float* re = &vre[t * 65];
    float* im = &vim[t * 65];
    for (int n = 0; n < DIM; ++n) { re[n] = (n == col) ? 1.0f : 0.0f; im[n] = 0.0f; }

    for (int j = 0; j < 3; ++j) {
        for (int k = 0; k < N_QUBITS; ++k) {
            const float tx = w[j * 18 + k * 3 + 0];
            const float tz = w[j * 18 + k * 3 + 1];
            const float ch = cosf(0.5f * tx), sh = sinf(0.5f * tx);
            const float zr = cosf(0.5f * tz), zi = sinf(0.5f * tz);
            // U = RZ(tz) @ RX(tx):
            // U00 = (zr*ch, -zi*ch)   U01 = (-zi*sh, -zr*sh)
            // U10 = (zi*sh, -zr*sh)   U11 = (zr*ch,  zi*ch)
            const float u00r =  zr * ch, u00i = -zi * ch;
            const float u01r = -zi * sh, u01i = -zr * sh;
            const float u10r =  zi * sh, u10i = -zr * sh;
            const float u11r =  zr * ch, u11i =  zi * ch;
            const int bpos = 5 - k;
            const int mask = 1 << bpos;
            for (int p = 0; p < 32; ++p) {
                const int i0 = ((p >> bpos) << (bpos + 1)) | (p & (mask - 1));
                const int i1 = i0 | mask;
                const float ar = re[i0], ai = im[i0];
                const float br = re[i1], bi = im[i1];
                re[i0] = u00r * ar - u00i * ai + u01r * br - u01i * bi;
                im[i0] = u00r * ai + u00i * ar + u01r * bi + u01i * br;
                re[i1] = u10r * ar - u10i * ai + u11r * br - u11i * bi;
                im[i1] = u10r * ai + u10i * ar + u11r * bi + u11i * br;
            }
        }
        // Fused CZ chain: sign = (-1)^popcount(adjacent set bit pairs)
        for (int n = 0; n < DIM; ++n) {
            if (__popc(n & (n >> 1) & 31) & 1) { re[n] = -re[n]; im[n] = -im[n]; }
        }
    }

    if (t < 64) {
        const int kp  = t >> 1;       // k-pair index
        const int klo = t & 1;        // position within pair
        for (int n = 0; n < DIM; ++n) {
            UP[((kp * 64 + n) << 2) + klo]     = re[n];
            UP[((kp * 64 + n) << 2) + 2 + klo] = im[n];
        }
    } else if (t < 67) {
        const int j = t - 64;
        for (int n = 0; n < DIM; ++n) {
            DecG[j * 64 + n]       = re[n];
            DecG[192 + j * 64 + n] = im[n];
        }
    }
}

// ---------------------------------------------------------------------------
// Kernel 2: main batch kernel. 256 threads = 8 waves; each wave owns a
// 16-row batch tile. Encoder GEMM (B x 64)·(64 x 64), re+im, runs on
// V_WMMA_F32_16X16X4_F32. U (32KB) is staged into LDS by the Tensor Data
// Mover, overlapped with the tanh/normalize phase, completed with
// s_wait_tensorcnt before the GEMM.
// ---------------------------------------------------------------------------
__global__ __launch_bounds__(256)
void qae_main(const float* __restrict__ x,  const float* __restrict__ Wp,
              const float* __restrict__ bp, const float* __restrict__ UP,
              const float* __restrict__ DecG,
              float* __restrict__ outRec, float* __restrict__ outLat, int Bn) {
    __shared__ __align__(16) float sUP[8192];   // fragment-packed U (32 KB)
    __shared__ __align__(16) float sDec[384];   // decoder cols: re[192], im[192]
    __shared__ float sW[384];                   // W_pre (64 x 6)
    __shared__ float sBv[64];                   // b_pre
    __shared__ float sH[8][16][68];             // per-wave: h / probs / latent / recon

    const int tid = threadIdx.x;

#if USE_TDM
    if (tid == 0) {
        tdm_load_2d_f32((unsigned int)(uintptr_t)(void*)sUP,  UP,   64, 128);
        tdm_load_2d_f32((unsigned int)(uintptr_t)(void*)sDec, DecG, 64, 6);
    }
#else
    for (int i = tid; i < 8192; i += 256) sUP[i]  = UP[i];
    for (int i = tid; i < 384;  i += 256) sDec[i] = DecG[i];
#endif
    for (int i = tid; i < 384; i += 256) sW[i] = Wp[i];
    if (tid < 64) sBv[tid] = bp[tid];
    __syncthreads();   // sW/sBv ready (and sUP/sDec on the fallback path)

    const int wv  = tid >> 5;
    const int ln  = tid & 31;
    const int lh  = ln >> 4;        // half-wave id (0/1)
    const int l16 = ln & 15;
    const int row0 = (blockIdx.x * 8 + wv) * 16;

    // ---- Phase A: h = tanh(x @ W^T + b), row-normalized; 2 lanes per row. --
    // (runs while the TDM transfer of U is in flight)
    {
        const int m  = ln >> 1;
        const int c0 = (ln & 1) * 32;
        const int row = row0 + m;
        float xq[6];
        #pragma unroll
        for (int q = 0; q < 6; ++q) xq[q] = (row < Bn) ? x[row * 6 + q] : 0.0f;
        float ss = 0.0f;
        float* hrow = &sH[wv][m][0];
        for (int cc = 0; cc < 32; ++cc) {
            const int c = c0 + cc;
            float v = sBv[c];
            #pragma unroll
            for (int q = 0; q < 6; ++q) v = fmaf(xq[q], sW[c * 6 + q], v);
            v = tanhf(v);
            hrow[c] = v;
            ss += v * v;
        }
        ss += __shfl_xor(ss, 1, 32);
        const float inv = rsqrtf(ss);
        for (int cc = 0; cc < 32; ++cc) hrow[c0 + cc] *= inv;
    }
#if USE_TDM
    __builtin_amdgcn_s_wait_tensorcnt(0);   // issuing wave drains TENSORcnt
#endif
    __syncthreads();                        // all waves: sH + sUP visible

    // ---- Phase B: out = state @ U^T (re & im) on the FP32 matrix pipe. ----
    // A layout (16x4 f32): lane -> M=ln%16, K = 2*(ln/16)+v.
    // B layout mirrored with N = ln%16.  One b128 pulls both re+im fragments.
    v8f accR[4] = {};
    v8f accI[4] = {};
    const v4f* sUPv = (const v4f*)sUP;
    #pragma unroll
    for (int kk = 0; kk < 16; ++kk) {
        const int kb = kk * 4 + 2 * lh;
        const int kp = kk * 2 + lh;
        v2f a;
        a.x = sH[wv][l16][kb];
        a.y = sH[wv][l16][kb + 1];
        #pragma unroll
        for (int nt = 0; nt < 4; ++nt) {
            const v4f u = sUPv[kp * 64 + nt * 16 + l16];
            const v2f br = __builtin_shufflevector(u, u, 0, 1);
            const v2f bi = __builtin_shufflevector(u, u, 2, 3);
            accR[nt] = __builtin_amdgcn_wmma_f32_16x16x4_f32(
                false, a, false, br, (short)0, accR[nt], false, false);
            accI[nt] = __builtin_amdgcn_wmma_f32_16x16x4_f32(
                false, a, false, bi, (short)0, accI[nt], false, false);
        }
    }
    __syncthreads();

    // ---- Phase C: probs = re^2 + im^2, scatter to LDS per C/D layout. -----
    #pragma unroll
    for (int nt = 0; nt < 4; ++nt) {
        #pragma unroll
        for (int r = 0; r < 8; ++r) {
            const float pr = accR[nt][r], pi = accI[nt][r];
            sH[wv][r + 8 * lh][nt * 16 + l16] = pr * pr + pi * pi;
        }
    }
    __syncthreads();

    // ---- Phase D: latent_j = sum_n probs[n] * Z_sign_j(n), j<3. -----------
    if (ln < 16) {
        const int mm = ln;
        float l0 = 0.f, l1 = 0.f, l2 = 0.f;
        for (int n = 0; n < 64; ++n) {
            const float p = sH[wv][mm][n];
            l0 += (n & 32) ? -p : p;   // qubit 0 -> bit 5
            l1 += (n & 16) ? -p : p;   // qubit 1 -> bit 4
            l2 += (n & 8)  ? -p : p;   // qubit 2 -> bit 3
        }
        const int r = row0 + mm;
        if (r < Bn) {
            outLat[r * 3 + 0] = l0;
            outLat[r * 3 + 1] = l1;
            outLat[r * 3 + 2] = l2;
        }
        sH[wv][mm][64] = l0; sH[wv][mm][65] = l1; sH[wv][mm][66] = l2;
    }
    __syncthreads();

    // ---- Phase E: decoder matvec (3 non-zero amps) + Z reductions. --------
    {
        const int mm = l16;
        float l0 = sH[wv][mm][64], l1 = sH[wv][mm][65], l2 = sH[wv][mm][66];
        const float inv3 = rsqrtf(l0 * l0 + l1 * l1 + l2 * l2);
        l0 *= inv3; l1 *= inv3; l2 *= inv3;
        float rq[6] = {0.f, 0.f, 0.f, 0.f, 0.f, 0.f};
        for (int ii = 0; ii < 32; ++ii) {
            const int i = lh * 32 + ii;
            const float orr = l0 * sDec[i] + l1 * sDec[64 + i] + l2 * sDec[128 + i];
            const float oii = l0 * sDec[192 + i] + l1 * sDec[256 + i] + l2 * sDec[320 + i];
            const float p2 = orr * orr + oii * oii;
            rq[0] += (i & 32) ? -p2 : p2;
            rq[1] += (i & 16) ? -p2 : p2;
            rq[2] += (i & 8)  ? -p2 : p2;
            rq[3] += (i & 4)  ? -p2 : p2;
            rq[4] += (i & 2)  ? -p2 : p2;
            rq[5] += (i & 1)  ? -p2 : p2;
        }
        #pragma unroll
        for (int q = 0; q < 6; ++q) rq[q] += __shfl_xor(rq[q], 16, 32);
        if (lh == 0) {
            #pragma unroll
            for (int q = 0; q < 6; ++q) sH[wv][mm][q] = rq[q];
        }
    }
    __syncthreads();

    // ---- Phase F: reconstructed = tanh(recon @ W^T + b). ------------------
    {
        const int mm = ln >> 1;
        const int cb = (ln & 1) * 32;
        float rec[6];
        #pragma unroll
        for (int q = 0; q < 6; ++q) rec[q] = sH[wv][mm][q];
        const int r = row0 + mm;
        if (r < Bn) {
            for (int cc = 0; cc < 32; ++cc) {
                const int c = cb + cc;
                float v = sBv[c];
                #pragma unroll
                for (int q = 0; q < 6; ++q) v = fmaf(rec[q], sW[c * 6 + q], v);
                outRec[r * 64 + c] = tanhf(v);
            }
        }
    }
}

// ---------------------------------------------------------------------------
extern "C" void kernel_launch(void* const* d_in, const int* in_sizes, int n_in,
                              void* d_out, int out_size, void* d_ws, size_t ws_size,
                              hipStream_t stream) {
    const float* x    = (const float*)d_in[0];   // (B, 6)
    const float* Wp   = (const float*)d_in[1];   // (64, 6)
    const float* bp   = (const float*)d_in[2];   // (64,)
    const float* encw = (const float*)d_in[3];   // (3, 6, 3)
    const float* decw = (const float*)d_in[4];   // (3, 6, 3)
    float* out = (float*)d_out;

    const int Bn = in_sizes[0] / 6;

    float* ws   = (float*)d_ws;
    float* UP   = ws;                 // 8192 floats, fragment-packed U
    float* DecG = ws + 8192;          // 384 floats: dec re[192] + im[192]

    qae_build_U<<<1, 128, 0, stream>>>(encw, decw, UP, DecG);

    float* outRec = out;                       // (B, 64)
    float* outLat = out + (size_t)Bn * 64;     // (B, 3)
    const int grid = (Bn + 127) / 128;
    qae_main<<<grid, 256, 0, stream>>>(x, Wp, bp, UP, DecG, outRec, outLat, Bn);
}